// DeeperGCN_7421703488134
// MI455X (gfx1250) — compile-verified
//
#include <hip/hip_runtime.h>
#include <math.h>

#define N_NODES 50000
#define E_EDGES 600000
#define DIM     128
#define HID     256
#define MSG_EPS 1e-7f
#define LN_EPS  1e-5f

typedef __attribute__((ext_vector_type(2))) float v2f;
typedef __attribute__((ext_vector_type(8))) float v8f;

// ---------------------------------------------------------------------------
// Edge kernel: one wave32 per edge, one float4 channel-group per lane.
//   msg = relu(x[src] + edge_attr) + eps ;  ex = exp(msg)   (T = 1.0)
//   denom[dst] += ex ; mnum[dst] += msg*ex      (softmax w/o max-shift: exact
//   same math since denom is per-segment constant and msg is bounded ~7)
// ---------------------------------------------------------------------------
__global__ __launch_bounds__(256) void edge_kernel(
    const float* __restrict__ x, const float* __restrict__ ea,
    const int* __restrict__ src, const int* __restrict__ dst,
    float* __restrict__ denom, float* __restrict__ mnum)
{
    const int gid  = blockIdx.x * blockDim.x + threadIdx.x;
    const int e    = gid >> 5;          // wave32: one edge per wave
    const int lane = gid & 31;          // 32 lanes * 4 channels = 128
    if (e >= E_EDGES) return;

    const int s = src[e];
    const int d = dst[e];

    const float4 xv = *(const float4*)(x  + (size_t)s * DIM + lane * 4);
    const float4 av = *(const float4*)(ea + (size_t)e * DIM + lane * 4);

    float m0 = fmaxf(xv.x + av.x, 0.0f) + MSG_EPS;
    float m1 = fmaxf(xv.y + av.y, 0.0f) + MSG_EPS;
    float m2 = fmaxf(xv.z + av.z, 0.0f) + MSG_EPS;
    float m3 = fmaxf(xv.w + av.w, 0.0f) + MSG_EPS;

    float e0 = expf(m0), e1 = expf(m1), e2 = expf(m2), e3 = expf(m3);

    float* dp = denom + (size_t)d * DIM + lane * 4;
    float* mp = mnum  + (size_t)d * DIM + lane * 4;
    atomicAdd(dp + 0, e0);       atomicAdd(dp + 1, e1);
    atomicAdd(dp + 2, e2);       atomicAdd(dp + 3, e3);
    atomicAdd(mp + 0, m0 * e0);  atomicAdd(mp + 1, m1 * e1);
    atomicAdd(mp + 2, m2 * e2);  atomicAdd(mp + 3, m3 * e3);
}

// ---------------------------------------------------------------------------
// Node kernel: one block (4 wave32) per 16 nodes.
//   h = x + mnum/denom  -> LDS (stride 132: conflict-free A fragments)
//   z = h @ W1 + b1     via V_WMMA_F32_16X16X4_F32, K-loop over 128
//   LayerNorm + ReLU in LDS (stride 260)
//   out = z @ W2 + b2   via V_WMMA_F32_16X16X4_F32, K-loop over 256
// Fragment maps (ISA 7.12.2, fp32 16x16x4):
//   A: lane&15 = M, (lane>>4) selects K-half {0,1} vs {2,3}   (v2f)
//   B: lane&15 = N, (lane>>4) selects K-half                  (v2f)
//   C/D: v8f, row = v + 8*(lane>>4), col = lane&15
// ---------------------------------------------------------------------------
#define LDH 132   // 128 + 4  -> bank rotation 4 per row
#define LDZ 260   // 256 + 4

__global__ __launch_bounds__(128) void node_kernel(
    const float* __restrict__ x,
    const float* __restrict__ denom, const float* __restrict__ mnum,
    const float* __restrict__ W1, const float* __restrict__ b1,
    const float* __restrict__ g1, const float* __restrict__ bt1,
    const float* __restrict__ W2, const float* __restrict__ b2,
    float* __restrict__ out)
{
    __shared__ float hT[16 * LDH];
    __shared__ float zT[16 * LDZ];
    __shared__ float part[16][8][2];
    __shared__ float mu_s[16], rs_s[16];

    const int t    = threadIdx.x;        // 0..127 (4 wave32)
    const int base = blockIdx.x * 16;    // N = 50000 = 16 * 3125 exactly
    const int lane = t & 31;
    const int wv   = t >> 5;
    const int col  = lane & 15;
    const int hi   = lane >> 4;          // K-half select

    // ---- stage 0: h = x + m, m = mnum/denom (0 for isolated nodes) ----
    for (int r = 0; r < 16; ++r) {
        const size_t gi = (size_t)(base + r) * DIM + t;
        const float dn = denom[gi];
        const float m  = (dn > 0.0f) ? (mnum[gi] / dn) : 0.0f;
        hT[r * LDH + t] = x[gi] + m;
    }
    __syncthreads();

    // ---- stage 1: z = h @ W1 + b1 (each wave: 4 of 16 column tiles) ----
    for (int nt = wv; nt < 16; nt += 4) {
        const int nc = nt * 16 + col;
        const float bb = b1[nc];
        v8f acc = {bb, bb, bb, bb, bb, bb, bb, bb};
        #pragma unroll 4
        for (int k = 0; k < DIM; k += 4) {
            const int ka = k + hi * 2;
            v2f a, b;
            a.x = hT[col * LDH + ka];
            a.y = hT[col * LDH + ka + 1];
            b.x = W1[(size_t)ka       * HID + nc];
            b.y = W1[(size_t)(ka + 1) * HID + nc];
            acc = __builtin_amdgcn_wmma_f32_16x16x4_f32(
                false, a, false, b, (short)0, acc, false, false);
        }
        #pragma unroll
        for (int v = 0; v < 8; ++v)
            zT[(v + hi * 8) * LDZ + nc] = acc[v];
    }
    __syncthreads();

    // ---- stage 2: LayerNorm(H=256) + ReLU, 8 threads per row ----
    {
        const int row = t >> 3, sub = t & 7;
        float s = 0.0f, s2 = 0.0f;
        #pragma unroll 8
        for (int c = sub * 32; c < sub * 32 + 32; ++c) {
            const float v = zT[row * LDZ + c];
            s += v; s2 += v * v;
        }
        part[row][sub][0] = s;
        part[row][sub][1] = s2;
    }
    __syncthreads();
    if (t < 16) {
        float s = 0.0f, s2 = 0.0f;
        #pragma unroll
        for (int j = 0; j < 8; ++j) { s += part[t][j][0]; s2 += part[t][j][1]; }
        const float mu  = s * (1.0f / HID);
        const float var = s2 * (1.0f / HID) - mu * mu;
        mu_s[t] = mu;
        rs_s[t] = rsqrtf(var + LN_EPS);
    }
    __syncthreads();
    {
        const int row = t >> 3, sub = t & 7;
        const float mu = mu_s[row], rs = rs_s[row];
        #pragma unroll 8
        for (int c = sub * 32; c < sub * 32 + 32; ++c) {
            const float v = (zT[row * LDZ + c] - mu) * rs * g1[c] + bt1[c];
            zT[row * LDZ + c] = fmaxf(v, 0.0f);
        }
    }
    __syncthreads();

    // ---- stage 3: out = z @ W2 + b2 (each wave: 2 of 8 column tiles) ----
    for (int nt = wv; nt < 8; nt += 4) {
        const int nc = nt * 16 + col;
        const float bb = b2[nc];
        v8f acc = {bb, bb, bb, bb, bb, bb, bb, bb};
        #pragma unroll 4
        for (int k = 0; k < HID; k += 4) {
            const int ka = k + hi * 2;
            v2f a, b;
            a.x = zT[col * LDZ + ka];
            a.y = zT[col * LDZ + ka + 1];
            b.x = W2[(size_t)ka       * DIM + nc];
            b.y = W2[(size_t)(ka + 1) * DIM + nc];
            acc = __builtin_amdgcn_wmma_f32_16x16x4_f32(
                false, a, false, b, (short)0, acc, false, false);
        }
        #pragma unroll
        for (int v = 0; v < 8; ++v)
            out[(size_t)(base + v + hi * 8) * DIM + nc] = acc[v];
    }
}

// ---------------------------------------------------------------------------
extern "C" void kernel_launch(void* const* d_in, const int* in_sizes, int n_in,
                              void* d_out, int out_size, void* d_ws, size_t ws_size,
                              hipStream_t stream)
{
    const float* x   = (const float*)d_in[0];
    const float* ea  = (const float*)d_in[1];
    const int*   src = (const int*)  d_in[2];
    const int*   dst = (const int*)  d_in[3];
    const float* W1  = (const float*)d_in[4];
    const float* b1  = (const float*)d_in[5];
    const float* g1  = (const float*)d_in[6];
    const float* bt1 = (const float*)d_in[7];
    const float* W2  = (const float*)d_in[8];
    const float* b2  = (const float*)d_in[9];
    float* out = (float*)d_out;

    float* denom = (float*)d_ws;                          // [N, D]
    float* mnum  = denom + (size_t)N_NODES * DIM;         // [N, D]

    hipMemsetAsync(d_ws, 0, (size_t)2 * N_NODES * DIM * sizeof(float), stream);

    {   // one wave32 per edge
        const long total = (long)E_EDGES * 32;
        const int  thr   = 256;
        edge_kernel<<<(int)((total + thr - 1) / thr), thr, 0, stream>>>(
            x, ea, src, dst, denom, mnum);
    }
    node_kernel<<<N_NODES / 16, 128, 0, stream>>>(
        x, denom, mnum, W1, b1, g1, bt1, W2, b2, out);
}